// GraphReversePool_53145925320769
// MI455X (gfx1250) — compile-verified
//
#include <hip/hip_runtime.h>
#include <stdint.h>

// Problem constants (from reference): x:[1024,25000] f32, v2c:[100000] i32,
// out:[1024,100000] f32.  out[b,v] = x[b, v2c[v]]  -- pure gather, 0 FLOPs.
// Roofline: ~512 MB min HBM traffic -> ~22 us @ 23.3 TB/s. Pure bandwidth.
#define NCLU  25000     // clusters per row (x row length)
#define NVERT 100000    // vertices (out row length)
#define TPB   256       // 8 wave32 per workgroup

typedef __attribute__((ext_vector_type(4))) float v4f;
typedef __attribute__((ext_vector_type(4))) int   v4i;

// Builtin signature (per hipcc diagnostic): (v4i addrspace(1)*, v4i addrspace(3)*, Ii, Ii)
typedef __attribute__((address_space(1))) v4i gas_v4i;   // global 16B chunk
typedef __attribute__((address_space(3))) v4i las_v4i;   // LDS 16B chunk

#if __has_builtin(__builtin_amdgcn_global_load_async_to_lds_b128)
#define HAVE_ASYNC_LDS 1
#else
#define HAVE_ASYNC_LDS 0
#warning "global_load_async_to_lds_b128 builtin not available; using VGPR-staged LDS copy fallback"
#endif

static __device__ __forceinline__ void wait_asynccnt0() {
#if __has_builtin(__builtin_amdgcn_s_wait_asynccnt)
    __builtin_amdgcn_s_wait_asynccnt(0);
#else
    asm volatile("s_wait_asynccnt 0" ::: "memory");
#endif
}

__global__ __launch_bounds__(TPB) void
GraphReversePool_gather_kernel(const float* __restrict__ x,
                               const int*   __restrict__ v2c,
                               float*       __restrict__ out,
                               int batch)
{
    // 100 KB of LDS: one full x row.  3 WGs / WGP (320 KB) -> staging of one
    // block overlaps the gather phase of its neighbors.
    extern __shared__ float smem[];

    const int b   = blockIdx.x;
    if (b >= batch) return;
    const int tid = threadIdx.x;

    const float* __restrict__ xrow = x + (size_t)b * NCLU;

    // ---- Stage x[b, :] (25000 f32 = 6250 x 16B) into LDS ----
#if HAVE_ASYNC_LDS
    {
        const int chunks = NCLU / 4;   // 6250 float4 chunks
        for (int j = tid; j < chunks; j += TPB) {
            __builtin_amdgcn_global_load_async_to_lds_b128(
                (gas_v4i*)const_cast<float*>(xrow + 4 * j),
                (las_v4i*)(smem + 4 * j),
                0, /*cpol=*/0);
        }
        wait_asynccnt0();          // s_wait_asynccnt 0: own ASYNCcnt drained
    }
#else
    {
        const v4f* __restrict__ src = (const v4f*)xrow;
        v4f* dst = (v4f*)smem;
        for (int j = tid; j < NCLU / 4; j += TPB)
            dst[j] = src[j];
    }
#endif
    __syncthreads();               // all waves' staging visible to the WG

    // ---- Gather: each out[b, v] is the LDS-staged x value at v2c[v]  (v2c int4-coalesced, out NT f4) ----
    const v4i* __restrict__ idx4 = (const v4i*)v2c;        // NVERT % 4 == 0
    v4f* __restrict__ out4 = (v4f*)(out + (size_t)b * NVERT);

    const int n4 = NVERT / 4;      // 25000 float4 outputs per row
    for (int i = tid; i < n4; i += TPB) {
        v4i c = idx4[i];           // coalesced b128 load, L2-resident (400 KB)
        v4f r;
        r.x = smem[c.x];           // ds_load_b32 x4: banked random access
        r.y = smem[c.y];
        r.z = smem[c.z];
        r.w = smem[c.w];
        // 409.6 MB write-once stream: non-temporal so it doesn't evict x/v2c.
        __builtin_nontemporal_store(r, out4 + i);
    }
}

extern "C" void kernel_launch(void* const* d_in, const int* in_sizes, int n_in,
                              void* d_out, int out_size, void* d_ws, size_t ws_size,
                              hipStream_t stream)
{
    const float* x   = (const float*)d_in[0];   // [batch, NCLU] f32
    const int*   v2c = (const int*)  d_in[1];   // [NVERT] i32
    float*       out = (float*)d_out;           // [batch, NVERT] f32

    const int batch = in_sizes[0] / NCLU;       // 1024 for this problem

    const size_t lds_bytes = (size_t)NCLU * sizeof(float);  // 100 KB / WG
    GraphReversePool_gather_kernel<<<dim3(batch), dim3(TPB), lds_bytes, stream>>>(
        x, v2c, out, batch);
}